// QueryTower_33062658245484
// MI455X (gfx1250) — compile-verified
//
#include <hip/hip_runtime.h>
#include <hip/hip_bf16.h>

// Problem constants (from reference)
#define BQ   4096
#define TQ   64
#define DIN  300
#define HQ   3
#define RQ   (BQ * TQ)          // 262144 rows of the projection GEMM

typedef __attribute__((ext_vector_type(2))) float v2f;
typedef __attribute__((ext_vector_type(8))) float v8f;

// ---------------------------------------------------------------------------
// Kernel 1: xb[t][b][n] = sum_k x[b,t,k] * W_ih[n,k] + (b_ih[n] + b_hh[n])
// GEMM [R=262144, K=300] x [300, 3(->16)] using V_WMMA_F32_16X16X4_F32.
// One wave = 16 rows. 8 waves / block. LDS holds W^T padded to 16 columns.
// ---------------------------------------------------------------------------
__global__ __launch_bounds__(256) void rnn_proj_wmma(
    const float* __restrict__ x,     // [B,T,DIN] row-major == [R, DIN]
    const float* __restrict__ Wih,   // [H, DIN]
    const float* __restrict__ bih,   // [H]
    const float* __restrict__ bhh,   // [H]
    float* __restrict__ xb)          // [T][B][H]  (time-major for kernel 2)
{
    __shared__ float WT[DIN * 16];   // WT[kk*16 + n] = (n<H) ? Wih[n,kk] : 0

    const int tid = threadIdx.x;
    for (int i = tid; i < DIN * 16; i += 256) {
        int kk = i >> 4, n = i & 15;
        WT[i] = (n < HQ) ? Wih[n * DIN + kk] : 0.0f;
    }
    __syncthreads();

    const int lane = tid & 31;
    const int wave = tid >> 5;
    const int nl   = lane & 15;      // A: row-in-tile, B/D: column N
    const int half = lane >> 4;      // A/B: which K half this lane feeds
    const long tileRow = ((long)blockIdx.x * 8 + wave) * 16;
    const float* __restrict__ xrow = x + (tileRow + nl) * DIN;

    v8f acc = {0.f, 0.f, 0.f, 0.f, 0.f, 0.f, 0.f, 0.f};

    // 37 pairs of K=4 chunks, permuted K ordering so each lane's float4 load
    // feeds exactly its own fragment slots (lanes<16: K0,K1 ; lanes>=16: K2,K3).
    for (int c = 0; c < 37; ++c) {
        const int kk0 = c * 8 + half * 4;
        const float4 a = *(const float4*)(xrow + kk0);

        v2f a1 = { a.x, a.y };
        v2f b1 = { WT[(kk0 + 0) * 16 + nl], WT[(kk0 + 1) * 16 + nl] };
        acc = __builtin_amdgcn_wmma_f32_16x16x4_f32(
                  false, a1, false, b1, (short)0, acc, false, false);

        v2f a2 = { a.z, a.w };
        v2f b2 = { WT[(kk0 + 2) * 16 + nl], WT[(kk0 + 3) * 16 + nl] };
        acc = __builtin_amdgcn_wmma_f32_16x16x4_f32(
                  false, a2, false, b2, (short)0, acc, false, false);
    }
    {   // remainder chunk: K = 296..299 (standard fragment mapping)
        const int kk0 = 296 + half * 2;
        const float2 a = *(const float2*)(xrow + kk0);
        v2f a1 = { a.x, a.y };
        v2f b1 = { WT[(kk0 + 0) * 16 + nl], WT[(kk0 + 1) * 16 + nl] };
        acc = __builtin_amdgcn_wmma_f32_16x16x4_f32(
                  false, a1, false, b1, (short)0, acc, false, false);
    }

    // D layout: VGPR v, lanes 0-15 -> (M=v, N=lane); lanes 16-31 -> (M=v+8).
    // Only columns N=0..2 are meaningful.
    if (nl < HQ) {
        const float bias = bih[nl] + bhh[nl];
#pragma unroll
        for (int v = 0; v < 8; ++v) {
            const long r = tileRow + v + 8 * half;      // global row = b*T + t
            const int  t = (int)(r & (TQ - 1));
            const long b = r >> 6;
            xb[((long)t * BQ + b) * HQ + nl] = acc[v] + bias;
        }
    }
}

// ---------------------------------------------------------------------------
// Kernel 2: masked tanh-RNN scan. One thread per batch element.
// h_j <- tanh(xb[t,b,j] + sum_k W_hh[j,k] * h_k) for t < lengths[b].
// xb already contains b_ih + b_hh.
// ---------------------------------------------------------------------------
__global__ __launch_bounds__(256) void rnn_scan(
    const float* __restrict__ xb,      // [T][B][H]
    const int*   __restrict__ lengths, // [B]
    const float* __restrict__ Whh,     // [H, H]
    float* __restrict__ out)           // [B, H]
{
    const int b = blockIdx.x * blockDim.x + threadIdx.x;
    if (b >= BQ) return;

    const float w00 = Whh[0], w01 = Whh[1], w02 = Whh[2];
    const float w10 = Whh[3], w11 = Whh[4], w12 = Whh[5];
    const float w20 = Whh[6], w21 = Whh[7], w22 = Whh[8];

    int len = lengths[b];
    len = len < 0 ? 0 : (len > TQ ? TQ : len);

    float h0 = 0.f, h1 = 0.f, h2 = 0.f;
    for (int t = 0; t < len; ++t) {
        const float* p = xb + ((long)t * BQ + b) * HQ;
        const float x0 = p[0], x1 = p[1], x2 = p[2];
        const float n0 = tanhf(fmaf(w00, h0, fmaf(w01, h1, fmaf(w02, h2, x0))));
        const float n1 = tanhf(fmaf(w10, h0, fmaf(w11, h1, fmaf(w12, h2, x1))));
        const float n2 = tanhf(fmaf(w20, h0, fmaf(w21, h1, fmaf(w22, h2, x2))));
        h0 = n0; h1 = n1; h2 = n2;
    }
    out[b * HQ + 0] = h0;
    out[b * HQ + 1] = h1;
    out[b * HQ + 2] = h2;
}

// ---------------------------------------------------------------------------
extern "C" void kernel_launch(void* const* d_in, const int* in_sizes, int n_in,
                              void* d_out, int out_size, void* d_ws, size_t ws_size,
                              hipStream_t stream)
{
    const float* x       = (const float*)d_in[0];   // [B,T,DIN]
    const int*   lengths = (const int*)  d_in[1];   // [B]
    const float* Wih     = (const float*)d_in[2];   // [H,DIN]
    const float* Whh     = (const float*)d_in[3];   // [H,H]
    const float* bih     = (const float*)d_in[4];   // [H]
    const float* bhh     = (const float*)d_in[5];   // [H]
    float*       out     = (float*)d_out;           // [B,H]
    float*       xb      = (float*)d_ws;            // [T][B][H] = 3.1 MB

    // 262144 rows / (16 rows/wave * 8 waves/block) = 2048 blocks
    rnn_proj_wmma<<<RQ / 128, 256, 0, stream>>>(x, Wih, bih, bhh, xb);
    rnn_scan<<<(BQ + 255) / 256, 256, 0, stream>>>(xb, lengths, Whh, out);
}